// MyDCGRUSTE_73065983640236
// MI455X (gfx1250) — compile-verified
//
#include <hip/hip_runtime.h>
#include <hip/hip_bf16.h>
#include <math.h>
#include <stdint.h>

// ---------------------------------------------------------------------------
// Problem constants (from reference)
// ---------------------------------------------------------------------------
#define Bz   4
#define Pz   12
#define Qz   12
#define Nz   2048
#define Dz   32
#define NMz  4            // K+1 Chebyshev terms
#define DAYSz 7
#define TODz  288
#define CDIM (Bz * 2 * Dz)   // 256 columns of the Chebyshev GEMMs
#define TWO_D (2 * Dz)       // 64

typedef float v2f __attribute__((ext_vector_type(2)));
typedef float v8f __attribute__((ext_vector_type(8)));

// ---------------------------------------------------------------------------
// CDNA5 async global->LDS copy (ASYNCcnt-tracked) + waits, via inline asm so
// the same source works on ROCm 7.2 clang-22 and the amdgpu-toolchain lane.
// LDS operand: low 32 bits of the generic pointer are the LDS byte address
// (ISA 10.2 aperture mapping).
// ---------------------------------------------------------------------------
__device__ __forceinline__ void async_copy_b128(const float* g, float* l) {
    asm volatile("global_load_async_to_lds_b128 %0, %1, off"
                 :
                 : "v"((uint32_t)(uintptr_t)l), "v"(g)
                 : "memory");
}
__device__ __forceinline__ void wait_async_le8() {
    asm volatile("s_wait_asynccnt 0x8" ::: "memory");
}
__device__ __forceinline__ void wait_async_le0() {
    asm volatile("s_wait_asynccnt 0x0" ::: "memory");
}

// ---------------------------------------------------------------------------
// SE embedding: se = relu(SE @ W_se1 + b1) @ W_se2 + b2     (N x D)
// ---------------------------------------------------------------------------
__global__ __launch_bounds__(32) void se_kernel(
    const float* __restrict__ SE, const float* __restrict__ W1,
    const float* __restrict__ b1, const float* __restrict__ W2,
    const float* __restrict__ b2, float* __restrict__ se_out)
{
    const int n = blockIdx.x;
    const int j = threadIdx.x;
    __shared__ float hid[Dz];
    float a = b1[j];
    #pragma unroll
    for (int i = 0; i < Dz; ++i) a += SE[n * Dz + i] * W1[i * Dz + j];
    hid[j] = fmaxf(a, 0.0f);
    __syncthreads();
    float o = b2[j];
    #pragma unroll
    for (int i = 0; i < Dz; ++i) o += hid[i] * W2[i * Dz + j];
    se_out[n * Dz + j] = o;
}

// ---------------------------------------------------------------------------
// TE embedding: one-hot rows of W_te1 are just two gathers.
// ---------------------------------------------------------------------------
__global__ __launch_bounds__(32) void te_kernel(
    const int* __restrict__ TE, const float* __restrict__ W1,
    const float* __restrict__ b1, const float* __restrict__ W2,
    const float* __restrict__ b2, float* __restrict__ te_out)
{
    const int row = blockIdx.x;          // b*(P+Q)+t
    const int j   = threadIdx.x;
    __shared__ float hid[Dz];
    const int day = TE[row * 2 + 0];
    const int tod = TE[row * 2 + 1];
    float a = W1[day * Dz + j] + W1[(DAYSz + tod) * Dz + j] + b1[j];
    hid[j] = fmaxf(a, 0.0f);
    __syncthreads();
    float o = b2[j];
    #pragma unroll
    for (int i = 0; i < Dz; ++i) o += hid[i] * W2[i * Dz + j];
    te_out[row * Dz + j] = o;
}

// ---------------------------------------------------------------------------
// Input MLP + STE add
// ---------------------------------------------------------------------------
__global__ __launch_bounds__(32) void xin_kernel(
    const float* __restrict__ X, const float* __restrict__ W1,
    const float* __restrict__ b1, const float* __restrict__ W2,
    const float* __restrict__ b2, const float* __restrict__ se,
    const float* __restrict__ te, float* __restrict__ xin)
{
    const int idx = blockIdx.x;          // ((b*P)+t)*N + n
    const int j   = threadIdx.x;
    const int n   = idx % Nz;
    const int bp  = idx / Nz;
    const int t   = bp % Pz;
    const int b   = bp / Pz;
    __shared__ float hid[Dz];
    const float xv = X[idx];
    hid[j] = fmaxf(xv * W1[j] + b1[j], 0.0f);
    __syncthreads();
    float o = b2[j];
    #pragma unroll
    for (int i = 0; i < Dz; ++i) o += hid[i] * W2[i * Dz + j];
    o += se[n * Dz + j] + te[(b * (Pz + Qz) + t) * Dz + j];
    xin[idx * Dz + j] = o;
}

// ---------------------------------------------------------------------------
// Pack cat = [xt, h] into GEMM layout x0[n][b*2D + f]   (N x 256)
// ---------------------------------------------------------------------------
__global__ __launch_bounds__(256) void pack_cat_kernel(
    const float* __restrict__ xin, const float* __restrict__ h,
    float* __restrict__ x0, int t)
{
    const int gid = blockIdx.x * blockDim.x + threadIdx.x;
    const int n   = gid / CDIM;
    const int col = gid % CDIM;
    const int b   = col / TWO_D;
    const int f   = col % TWO_D;
    float v;
    if (f < Dz) v = xin[(((b * Pz + t) * Nz) + n) * Dz + f];
    else        v = h[((b * Nz) + n) * Dz + (f - Dz)];
    x0[gid] = v;
}

// Pack cat2 = [xt, r*h] (r taken from ru[..., :D])
__global__ __launch_bounds__(256) void pack_cat2_kernel(
    const float* __restrict__ xin, const float* __restrict__ h,
    const float* __restrict__ ru, float* __restrict__ x0, int t)
{
    const int gid = blockIdx.x * blockDim.x + threadIdx.x;
    const int n   = gid / CDIM;
    const int col = gid % CDIM;
    const int b   = col / TWO_D;
    const int f   = col % TWO_D;
    float v;
    if (f < Dz) {
        v = xin[(((b * Pz + t) * Nz) + n) * Dz + f];
    } else {
        const int bn = b * Nz + n;
        const float r = ru[bn * TWO_D + (f - Dz)];
        v = r * h[bn * Dz + (f - Dz)];
    }
    x0[gid] = v;
}

// ---------------------------------------------------------------------------
// Chebyshev GEMM (fp32 WMMA, async-LDS staged):  Y = alpha*(L @ X) + beta*Z
//   L : 2048 x 2048,  X/Z/Y : 2048 x 256
// Block = 128 threads (4 waves), block tile 64x64, wave tile 32x32 (2x2 WMMA).
// K staged in 32-wide chunks, double-buffered via global_load_async_to_lds_b128
// + s_wait_asynccnt, WMMA fragments served from LDS.
// grid = (2048/64, 256/64) = (32, 4)
// ---------------------------------------------------------------------------
#define MB 64            // rows per block tile
#define CB 64            // cols per block tile
#define KC 32            // K chunk
#define A_STRIDE 36      // padded (row stride 144 B: 16B-aligned, bank-spread)
#define NCHUNK (Nz / KC) // 64

__global__ __launch_bounds__(128) void cheb_gemm_wmma(
    const float* __restrict__ Lm, const float* __restrict__ Xm,
    const float* __restrict__ Zm, float* __restrict__ Ym,
    float alpha, float beta)
{
    __shared__ float As[2][MB * A_STRIDE];   // 2 x 9216 B
    __shared__ float Bs[2][KC * CB];         // 2 x 8192 B

    const int tid  = threadIdx.x;
    const int wave = tid >> 5;
    const int lane = tid & 31;
    const int half = lane >> 4;      // which K/M half this lane holds
    const int l16  = lane & 15;
    const int wm   = wave >> 1;      // 0..1 -> row sub-tile
    const int wc   = wave & 1;       // 0..1 -> col sub-tile

    const int rowBlk = blockIdx.x * MB;
    const int colBlk = blockIdx.y * CB;

    v8f acc00 = {0.f,0.f,0.f,0.f,0.f,0.f,0.f,0.f};
    v8f acc01 = acc00, acc10 = acc00, acc11 = acc00;

    // 8 async b128 copies per thread per chunk: 4 for A (64x32), 4 for B (32x64)
    auto issue_chunk = [&](int k0, int buf) {
        #pragma unroll
        for (int j = 0; j < 4; ++j) {
            const int i = tid + 128 * j;        // 0..511
            const int r = i >> 3;               // 8 x b128 per 32-float row
            const int q = (i & 7) << 2;         // float offset in row
            async_copy_b128(Lm + (size_t)(rowBlk + r) * Nz + k0 + q,
                            &As[buf][r * A_STRIDE + q]);
        }
        #pragma unroll
        for (int j = 0; j < 4; ++j) {
            const int i = tid + 128 * j;        // 0..511
            const int r = i >> 4;               // 16 x b128 per 64-float row
            const int q = (i & 15) << 2;
            async_copy_b128(Xm + (size_t)(k0 + r) * CDIM + colBlk + q,
                            &Bs[buf][r * CB + q]);
        }
    };

    issue_chunk(0, 0);

    #pragma unroll 1
    for (int c = 0; c < NCHUNK; ++c) {
        const int cur = c & 1;
        if (c + 1 < NCHUNK) {
            issue_chunk((c + 1) * KC, cur ^ 1);
            wait_async_le8();       // chunk c's 8 copies complete (in-order)
        } else {
            wait_async_le0();
        }
        __syncthreads();            // all waves' copies for chunk c visible

        const float* Ab = &As[cur][0];
        const float* Bb = &Bs[cur][0];
        #pragma unroll
        for (int kk = 0; kk < KC; kk += 4) {
            const int ka = kk + half * 2;
            v2f a0, a1, b0, b1;
            const float* ap = Ab + (wm * 32 + l16) * A_STRIDE + ka;
            a0.x = ap[0];                 a0.y = ap[1];
            a1.x = ap[16 * A_STRIDE];     a1.y = ap[16 * A_STRIDE + 1];
            const float* bp = Bb + ka * CB + wc * 32 + l16;
            b0.x = bp[0];                 b0.y = bp[CB];
            b1.x = bp[16];                b1.y = bp[CB + 16];

            acc00 = __builtin_amdgcn_wmma_f32_16x16x4_f32(
                        false, a0, false, b0, (short)0, acc00, false, false);
            acc01 = __builtin_amdgcn_wmma_f32_16x16x4_f32(
                        false, a0, false, b1, (short)0, acc01, false, false);
            acc10 = __builtin_amdgcn_wmma_f32_16x16x4_f32(
                        false, a1, false, b0, (short)0, acc10, false, false);
            acc11 = __builtin_amdgcn_wmma_f32_16x16x4_f32(
                        false, a1, false, b1, (short)0, acc11, false, false);
        }
        __syncthreads();            // reads done before buffer is overwritten
    }

    // C/D layout: VGPR r -> row (r + 8*half), col = l16
    const int row0 = rowBlk + wm * 32;
    const int col0 = colBlk + wc * 32;
    #pragma unroll
    for (int r = 0; r < 8; ++r) {
        const int m0 = row0 + r + half * 8;
        const int m1 = m0 + 16;
        const int c0 = col0 + l16;
        const int c1 = c0 + 16;
        size_t i00 = (size_t)m0 * CDIM + c0;
        size_t i01 = (size_t)m0 * CDIM + c1;
        size_t i10 = (size_t)m1 * CDIM + c0;
        size_t i11 = (size_t)m1 * CDIM + c1;
        Ym[i00] = alpha * acc00[r] + beta * Zm[i00];
        Ym[i01] = alpha * acc01[r] + beta * Zm[i01];
        Ym[i10] = alpha * acc10[r] + beta * Zm[i10];
        Ym[i11] = alpha * acc11[r] + beta * Zm[i11];
    }
}

// ---------------------------------------------------------------------------
// Gate projection: ru = sigmoid(xs @ W_gate + b_gate)   (B,N,2D)
// ---------------------------------------------------------------------------
__global__ __launch_bounds__(64) void gate_kernel(
    const float* __restrict__ x0, const float* __restrict__ x1,
    const float* __restrict__ x2, const float* __restrict__ x3,
    const float* __restrict__ Wg, const float* __restrict__ bg,
    float* __restrict__ ru)
{
    const int bn = blockIdx.x;           // b*N + n
    const int b  = bn / Nz;
    const int n  = bn % Nz;
    const int j  = threadIdx.x;          // 0..63
    __shared__ float row[NMz * TWO_D];   // 256
    const float* xs[NMz] = {x0, x1, x2, x3};
    for (int k = j; k < NMz * TWO_D; k += 64) {
        const int m = k / TWO_D;
        const int f = k % TWO_D;
        row[k] = xs[m][(size_t)n * CDIM + b * TWO_D + f];
    }
    __syncthreads();
    float acc = bg[j];
    #pragma unroll 4
    for (int k = 0; k < NMz * TWO_D; ++k)
        acc += row[k] * Wg[k * TWO_D + j];
    ru[(size_t)bn * TWO_D + j] = 1.0f / (1.0f + expf(-acc));
}

// ---------------------------------------------------------------------------
// Candidate projection + GRU update
// ---------------------------------------------------------------------------
__global__ __launch_bounds__(32) void cand_update_kernel(
    const float* __restrict__ x0, const float* __restrict__ x1,
    const float* __restrict__ x2, const float* __restrict__ x3,
    const float* __restrict__ Wc, const float* __restrict__ bc,
    const float* __restrict__ ru, float* __restrict__ h)
{
    const int bn = blockIdx.x;
    const int b  = bn / Nz;
    const int n  = bn % Nz;
    const int f  = threadIdx.x;          // 0..31
    __shared__ float row[NMz * TWO_D];
    const float* xs[NMz] = {x0, x1, x2, x3};
    for (int k = f; k < NMz * TWO_D; k += 32) {
        const int m = k / TWO_D;
        const int ff = k % TWO_D;
        row[k] = xs[m][(size_t)n * CDIM + b * TWO_D + ff];
    }
    __syncthreads();
    float acc = bc[f];
    #pragma unroll 4
    for (int k = 0; k < NMz * TWO_D; ++k)
        acc += row[k] * Wc[k * Dz + f];
    const float c  = tanhf(acc);
    const float u  = ru[(size_t)bn * TWO_D + Dz + f];
    const float ho = h[(size_t)bn * Dz + f];
    h[(size_t)bn * Dz + f] = u * ho + (1.0f - u) * c;
}

// ---------------------------------------------------------------------------
// Output head: y[b][q][n]
// ---------------------------------------------------------------------------
__global__ __launch_bounds__(32) void out_kernel(
    const float* __restrict__ h, const float* __restrict__ W1,
    const float* __restrict__ b1, const float* __restrict__ W2,
    const float* __restrict__ b2, float* __restrict__ y)
{
    const int bn = blockIdx.x;
    const int b  = bn / Nz;
    const int n  = bn % Nz;
    const int j  = threadIdx.x;
    __shared__ float hid[Dz];
    float a = b1[j];
    #pragma unroll
    for (int i = 0; i < Dz; ++i) a += h[(size_t)bn * Dz + i] * W1[i * Dz + j];
    hid[j] = fmaxf(a, 0.0f);
    __syncthreads();
    if (j < Qz) {
        float o = b2[j];
        #pragma unroll
        for (int i = 0; i < Dz; ++i) o += hid[i] * W2[i * Qz + j];
        y[((size_t)b * Qz + j) * Nz + n] = o;
    }
}

// ---------------------------------------------------------------------------
// Host orchestration
// ---------------------------------------------------------------------------
extern "C" void kernel_launch(void* const* d_in, const int* in_sizes, int n_in,
                              void* d_out, int out_size, void* d_ws, size_t ws_size,
                              hipStream_t stream) {
    const float* X      = (const float*)d_in[0];
    const int*   TE     = (const int*)  d_in[1];
    const float* L      = (const float*)d_in[2];
    const float* SE     = (const float*)d_in[3];
    const float* W_se1  = (const float*)d_in[4];
    const float* b_se1  = (const float*)d_in[5];
    const float* W_se2  = (const float*)d_in[6];
    const float* b_se2  = (const float*)d_in[7];
    const float* W_te1  = (const float*)d_in[8];
    const float* b_te1  = (const float*)d_in[9];
    const float* W_te2  = (const float*)d_in[10];
    const float* b_te2  = (const float*)d_in[11];
    const float* W_in1  = (const float*)d_in[12];
    const float* b_in1  = (const float*)d_in[13];
    const float* W_in2  = (const float*)d_in[14];
    const float* b_in2  = (const float*)d_in[15];
    const float* W_gate = (const float*)d_in[16];
    const float* b_gate = (const float*)d_in[17];
    const float* W_cand = (const float*)d_in[18];
    const float* b_cand = (const float*)d_in[19];
    const float* W_out1 = (const float*)d_in[20];
    const float* b_out1 = (const float*)d_in[21];
    const float* W_out2 = (const float*)d_in[22];
    const float* b_out2 = (const float*)d_in[23];
    float* out = (float*)d_out;

    // Workspace carve-up (floats)
    float* ws  = (float*)d_ws;
    float* se  = ws;                                 // N*D
    float* te  = se  + (size_t)Nz * Dz;              // B*(P+Q)*D
    float* xin = te  + (size_t)Bz * (Pz + Qz) * Dz;  // B*P*N*D
    float* h   = xin + (size_t)Bz * Pz * Nz * Dz;    // B*N*D
    float* ru  = h   + (size_t)Bz * Nz * Dz;         // B*N*2D
    float* x0  = ru  + (size_t)Bz * Nz * TWO_D;      // N*256
    float* x1  = x0  + (size_t)Nz * CDIM;
    float* x2  = x1  + (size_t)Nz * CDIM;
    float* x3  = x2  + (size_t)Nz * CDIM;

    // h0 = 0
    hipMemsetAsync(h, 0, (size_t)Bz * Nz * Dz * sizeof(float), stream);

    // Embeddings + input MLP
    se_kernel<<<Nz, 32, 0, stream>>>(SE, W_se1, b_se1, W_se2, b_se2, se);
    te_kernel<<<Bz * (Pz + Qz), 32, 0, stream>>>(TE, W_te1, b_te1, W_te2, b_te2, te);
    xin_kernel<<<Bz * Pz * Nz, 32, 0, stream>>>(X, W_in1, b_in1, W_in2, b_in2,
                                                se, te, xin);

    const dim3 ggrid(Nz / MB, CDIM / CB);   // (32, 4)
    const int packBlocks = (Nz * CDIM) / 256;

    for (int t = 0; t < Pz; ++t) {
        // ---- gate path ----
        pack_cat_kernel<<<packBlocks, 256, 0, stream>>>(xin, h, x0, t);
        cheb_gemm_wmma<<<ggrid, 128, 0, stream>>>(L, x0, x0, x1, 1.0f,  0.0f);
        cheb_gemm_wmma<<<ggrid, 128, 0, stream>>>(L, x1, x0, x2, 2.0f, -1.0f);
        cheb_gemm_wmma<<<ggrid, 128, 0, stream>>>(L, x2, x1, x3, 2.0f, -1.0f);
        gate_kernel<<<Bz * Nz, 64, 0, stream>>>(x0, x1, x2, x3, W_gate, b_gate, ru);

        // ---- candidate path ----
        pack_cat2_kernel<<<packBlocks, 256, 0, stream>>>(xin, h, ru, x0, t);
        cheb_gemm_wmma<<<ggrid, 128, 0, stream>>>(L, x0, x0, x1, 1.0f,  0.0f);
        cheb_gemm_wmma<<<ggrid, 128, 0, stream>>>(L, x1, x0, x2, 2.0f, -1.0f);
        cheb_gemm_wmma<<<ggrid, 128, 0, stream>>>(L, x2, x1, x3, 2.0f, -1.0f);
        cand_update_kernel<<<Bz * Nz, 32, 0, stream>>>(x0, x1, x2, x3, W_cand,
                                                       b_cand, ru, h);
    }

    out_kernel<<<Bz * Nz, 32, 0, stream>>>(h, W_out1, b_out1, W_out2, b_out2, out);
}